// ComputeAttention_59880434041543
// MI455X (gfx1250) — compile-verified
//
#include <hip/hip_runtime.h>
#include <float.h>

// Talking-heads causal attention, fp32, MI455X (gfx1250, wave32).
// d_out = out [B,H,S,D] then attn [B,H,S,S], concatenated.
// Pass 1: softmax stats only. Pass 2: recompute scores, normalize, post-mix,
// write attn once, AV. K/V tiles staged with async global->LDS (ASYNCcnt).

#define B_ 2
#define H_ 8
#define S_ 2048
#define D_ 64
#define SCALE_ 0.125f
#define NEGMAX_ (-3.4028234663852886e+38f)

typedef float v2f __attribute__((ext_vector_type(2)));
typedef float v8f __attribute__((ext_vector_type(8)));
typedef int v4i __attribute__((ext_vector_type(4)));

#if defined(__has_builtin)
#if __has_builtin(__builtin_amdgcn_global_load_async_to_lds_b128)
#define USE_ASYNC_LDS 1
#endif
#endif

__device__ __forceinline__ v8f wmma_f32_16x16x4(v2f a, v2f b, v8f c) {
  // (neg_a, A, neg_b, B, c_mod, C, reuse_a, reuse_b)
  return __builtin_amdgcn_wmma_f32_16x16x4_f32(false, a, false, b, (short)0, c,
                                               false, false);
}

// Stage a 16x64 f32 tile (4 KB) from global to LDS; 32 lanes x 8 x b128.
__device__ __forceinline__ void stage_tile(const float* __restrict__ gsrc,
                                           float* lds_dst, int lane) {
#ifdef USE_ASYNC_LDS
  typedef __attribute__((address_space(1))) v4i* gp_t;
  typedef __attribute__((address_space(3))) v4i* lp_t;
  gp_t g = (gp_t)(gsrc + lane * 4);     // 16 B per lane
  lp_t l = (lp_t)(lds_dst + lane * 4);
#pragma unroll
  for (int t = 0; t < 8; ++t)
    __builtin_amdgcn_global_load_async_to_lds_b128(g + t * 32, l + t * 32, 0,
                                                   0);
#else
  const float4* src = (const float4*)gsrc;
  float4* dst = (float4*)lds_dst;
#pragma unroll
  for (int t = 0; t < 8; ++t) dst[lane + 32 * t] = src[lane + 32 * t];
#endif
}

__device__ __forceinline__ void stage_wait() {
#ifdef USE_ASYNC_LDS
#if __has_builtin(__builtin_amdgcn_s_wait_asynccnt)
  __builtin_amdgcn_s_wait_asynccnt(0);
#else
  asm volatile("s_wait_asynccnt 0" ::: "memory");
#endif
#endif
}

// ---------------------------------------------------------------------------
// Kernel 1: streaming softmax stats (m, l) of the pre-mixed masked scores.
// Grid: B * S/16 workgroups, 256 threads (8 waves; wave w = head w).
// ---------------------------------------------------------------------------
__global__ __launch_bounds__(256) void th_stats(
    const float* __restrict__ q, const float* __restrict__ k,
    const unsigned char* __restrict__ mask, const float* __restrict__ pre_w,
    float* __restrict__ wsm, float* __restrict__ wsl) {
  __shared__ float Ks[H_][16 * D_];   // 32 KB
  __shared__ float mixd[H_][8 * 32];  // 8 KB

  const int tid = threadIdx.x;
  const int h = tid >> 5;
  const int lane = tid & 31;
  const int half = lane >> 4;
  const int N = lane & 15;

  const int nIt = S_ / 16;
  const int b = blockIdx.x / nIt;
  const int i0 = (blockIdx.x % nIt) * 16;

  float pw[H_];
#pragma unroll
  for (int hh = 0; hh < H_; ++hh) pw[hh] = pre_w[hh * H_ + h];

  // Q tile of head h as 16 A-fragments (K=4 each)
  v2f qa[16];
  {
    const float* qrow =
        q + (((size_t)(b * H_ + h)) * S_ + (size_t)(i0 + N)) * D_;
#pragma unroll
    for (int s = 0; s < 16; ++s) qa[s] = *(const v2f*)(qrow + 4 * s + 2 * half);
  }

  float m_r[8], l_r[8];
#pragma unroll
  for (int r = 0; r < 8; ++r) { m_r[r] = NEGMAX_; l_r[r] = 0.0f; }

  const int jt_end = (i0 + 15) / 16;
  for (int jt = 0; jt <= jt_end; ++jt) {
    const int j0 = jt * 16;
    stage_tile(k + (((size_t)(b * H_ + h)) * S_ + (size_t)j0) * D_, &Ks[h][0],
               lane);
    stage_wait();
    __syncthreads();

    // S_h = Q_h K_h^T; two independent 8-deep WMMA chains for ILP
    v8f acc0 = {}, acc1 = {};
#pragma unroll
    for (int s = 0; s < 8; ++s) {
      const int kk = 4 * s + 2 * half;
      v2f bf = *(const v2f*)&Ks[h][N * D_ + kk];
      acc0 = wmma_f32_16x16x4(qa[s], bf, acc0);
    }
#pragma unroll
    for (int s = 8; s < 16; ++s) {
      const int kk = 4 * s + 2 * half;
      v2f bf = *(const v2f*)&Ks[h][N * D_ + kk];
      acc1 = wmma_f32_16x16x4(qa[s], bf, acc1);
    }
#pragma unroll
    for (int r = 0; r < 8; ++r)
      mixd[h][r * 32 + lane] = (acc0[r] + acc1[r]) * SCALE_;
    __syncthreads();

    const int j = j0 + N;
    const bool mok = (mask[b * S_ + j] != 0);
#pragma unroll
    for (int r = 0; r < 8; ++r) {
      float x = 0.0f;
#pragma unroll
      for (int hh = 0; hh < H_; ++hh) x += pw[hh] * mixd[hh][r * 32 + lane];
      const int i = i0 + r + 8 * half;
      if (j > i || !mok) x = NEGMAX_;

      float rmax = x;
#pragma unroll
      for (int off = 8; off >= 1; off >>= 1)
        rmax = fmaxf(rmax, __shfl_xor(rmax, off));
      const float mn = fmaxf(m_r[r], rmax);
      float rsum = __expf(x - mn);
#pragma unroll
      for (int off = 8; off >= 1; off >>= 1) rsum += __shfl_xor(rsum, off);
      l_r[r] = l_r[r] * __expf(m_r[r] - mn) + rsum;
      m_r[r] = mn;
    }
    __syncthreads();  // everyone done with mixd/Ks before next tile
  }

  if (N == 0) {
#pragma unroll
    for (int r = 0; r < 8; ++r) {
      const size_t idx =
          ((size_t)(b * H_ + h)) * S_ + (size_t)(i0 + r + 8 * half);
      wsm[idx] = m_r[r];
      wsl[idx] = l_r[r];
    }
  }
}

// ---------------------------------------------------------------------------
// Kernel 2: recompute scores, normalize, post-mix, write attn once, AV -> out.
// ---------------------------------------------------------------------------
__global__ __launch_bounds__(256) void th_fused(
    const float* __restrict__ q, const float* __restrict__ k,
    const float* __restrict__ vmat, const unsigned char* __restrict__ mask,
    const float* __restrict__ pre_w, const float* __restrict__ post_w,
    const float* __restrict__ wsm, const float* __restrict__ wsl,
    float* __restrict__ attn, float* __restrict__ out) {
  __shared__ float Ks[H_][16 * D_];   // 32 KB
  __shared__ float Vs[H_][16 * D_];   // 32 KB
  __shared__ float mixd[H_][8 * 32];  // 8 KB: raw dots exchange
  __shared__ float mixp[H_][8 * 32];  // 8 KB: probability exchange
  __shared__ float aS[H_][16 * 16];   // 8 KB: mixed tile, (M,N) for A-frags

  const int tid = threadIdx.x;
  const int h = tid >> 5;  // wave id == head id == mixed-head id
  const int lane = tid & 31;
  const int half = lane >> 4;
  const int N = lane & 15;

  const int nIt = S_ / 16;
  const int b = blockIdx.x / nIt;
  const int i0 = (blockIdx.x % nIt) * 16;

  float pwpre[H_], pwpost[H_];
#pragma unroll
  for (int hh = 0; hh < H_; ++hh) {
    pwpre[hh] = pre_w[hh * H_ + h];
    pwpost[hh] = post_w[hh * H_ + h];
  }

  v2f qa[16];
  {
    const float* qrow =
        q + (((size_t)(b * H_ + h)) * S_ + (size_t)(i0 + N)) * D_;
#pragma unroll
    for (int s = 0; s < 16; ++s) qa[s] = *(const v2f*)(qrow + 4 * s + 2 * half);
  }

  float mg[8], rl[8];
#pragma unroll
  for (int r = 0; r < 8; ++r) {
    const size_t idx =
        ((size_t)(b * H_ + h)) * S_ + (size_t)(i0 + r + 8 * half);
    mg[r] = wsm[idx];
    rl[r] = 1.0f / wsl[idx];
  }

  v8f o[4] = {};
  float* ab = attn + ((size_t)(b * H_ + h)) * S_ * (size_t)S_;

  const int jt_end = (i0 + 15) / 16;
  for (int jt = 0; jt <= jt_end; ++jt) {
    const int j0 = jt * 16;
    // stage K and V tiles for head h (async global->LDS, ASYNCcnt tracked)
    stage_tile(k + (((size_t)(b * H_ + h)) * S_ + (size_t)j0) * D_, &Ks[h][0],
               lane);
    stage_tile(vmat + (((size_t)(b * H_ + h)) * S_ + (size_t)j0) * D_,
               &Vs[h][0], lane);
    stage_wait();
    __syncthreads();  // (1) staging visible

    // recompute S_h = Q_h K_h^T
    v8f acc0 = {}, acc1 = {};
#pragma unroll
    for (int s = 0; s < 8; ++s) {
      const int kk = 4 * s + 2 * half;
      v2f bf = *(const v2f*)&Ks[h][N * D_ + kk];
      acc0 = wmma_f32_16x16x4(qa[s], bf, acc0);
    }
#pragma unroll
    for (int s = 8; s < 16; ++s) {
      const int kk = 4 * s + 2 * half;
      v2f bf = *(const v2f*)&Ks[h][N * D_ + kk];
      acc1 = wmma_f32_16x16x4(qa[s], bf, acc1);
    }
#pragma unroll
    for (int r = 0; r < 8; ++r)
      mixd[h][r * 32 + lane] = (acc0[r] + acc1[r]) * SCALE_;
    __syncthreads();  // (2) dots published

    // premix + mask + normalize -> probabilities, publish
    const int j = j0 + N;
    const bool mok = (mask[b * S_ + j] != 0);
#pragma unroll
    for (int r = 0; r < 8; ++r) {
      float x = 0.0f;
#pragma unroll
      for (int hh = 0; hh < H_; ++hh) x += pwpre[hh] * mixd[hh][r * 32 + lane];
      const int i = i0 + r + 8 * half;
      if (j > i || !mok) x = NEGMAX_;
      mixp[h][r * 32 + lane] = __expf(x - mg[r]) * rl[r];
    }
    __syncthreads();  // (3) probabilities published

    // post-softmax mix; write final attn; stash for A-fragments
    float a[8];
#pragma unroll
    for (int r = 0; r < 8; ++r) {
      float vv = 0.0f;
#pragma unroll
      for (int hh = 0; hh < H_; ++hh) vv += pwpost[hh] * mixp[hh][r * 32 + lane];
      a[r] = vv;
      ab[(size_t)(i0 + r + 8 * half) * S_ + j] = vv;
      aS[h][(r + 8 * half) * 16 + N] = vv;
    }
    __syncthreads();  // (4) aS visible

    // o += a(16x16) @ V(16x64)
#pragma unroll
    for (int c = 0; c < 4; ++c) {
#pragma unroll
      for (int s = 0; s < 4; ++s) {
        const int kk = 4 * s + 2 * half;
        v2f af = *(const v2f*)&aS[h][N * 16 + kk];
        v2f bf;
        bf.x = Vs[h][kk * D_ + 16 * c + N];
        bf.y = Vs[h][(kk + 1) * D_ + 16 * c + N];
        o[c] = wmma_f32_16x16x4(af, bf, o[c]);
      }
    }
    __syncthreads();  // (5) AV reads done before next staging overwrites LDS
  }

  // zero-fill the causal-dead region: rows [i0,i0+16), cols [i0+16, S)
  {
    const int jstart = i0 + 16;
    if (jstart < S_) {
      const int n4 = (S_ - jstart) >> 2;  // float4 count per row
      const float4 z4 = {0.0f, 0.0f, 0.0f, 0.0f};
      for (int rm = 0; rm < 16; ++rm) {
        float4* rowp = (float4*)(ab + (size_t)(i0 + rm) * S_ + jstart);
        for (int t = lane; t < n4; t += 32) rowp[t] = z4;
      }
    }
  }

  // write out[b,h,i0+M,:]
#pragma unroll
  for (int c = 0; c < 4; ++c) {
#pragma unroll
    for (int r = 0; r < 8; ++r) {
      out[(((size_t)(b * H_ + h)) * S_ + (size_t)(i0 + r + 8 * half)) * D_ +
          16 * c + N] = o[c][r];
    }
  }
}

// ---------------------------------------------------------------------------
extern "C" void kernel_launch(void* const* d_in, const int* in_sizes, int n_in,
                              void* d_out, int out_size, void* d_ws,
                              size_t ws_size, hipStream_t stream) {
  const float* q = (const float*)d_in[0];
  const float* k = (const float*)d_in[1];
  const float* v = (const float*)d_in[2];
  const unsigned char* mask = (const unsigned char*)d_in[3];  // jnp bool = 1B
  const float* pre_w = (const float*)d_in[4];
  const float* post_w = (const float*)d_in[5];

  float* out = (float*)d_out;                     // [B,H,S,D]
  float* attn = out + (size_t)B_ * H_ * S_ * D_;  // [B,H,S,S]

  float* wsm = (float*)d_ws;                      // [B*H*S] row max
  float* wsl = wsm + (size_t)B_ * H_ * S_;        // [B*H*S] row sum-exp

  dim3 grid(B_ * (S_ / 16));
  dim3 block(256);
  th_stats<<<grid, block, 0, stream>>>(q, k, mask, pre_w, wsm, wsl);
  th_fused<<<grid, block, 0, stream>>>(q, k, v, mask, pre_w, post_w, wsm, wsl,
                                       attn, out);
}